// SpatialQAC_90821378441469
// MI455X (gfx1250) — compile-verified
//
#include <hip/hip_runtime.h>

// CDNA5 / gfx1250: wave32, WMMA f32 16x16x4, LDS-tiled stencil with
// GLOBAL_LOAD_ASYNC_TO_LDS halo fill (ASYNCcnt), 50 ping-pong step launches
// (state is 32MB -> fully L2-resident on MI455X's 192MB L2).

typedef float v2f __attribute__((ext_vector_type(2)));
typedef float v8f __attribute__((ext_vector_type(8)));

#define H_    512
#define W_    512
#define CCH   16
#define TH    16
#define TW    16
#define HT    18              // TH + 2 halo
#define PLANE (HT * HT)       // 324 floats per channel plane
#define DT_   0.05f
#define DIFF_ 0.1f
#define BETA_ 10.0f

// Async memory->LDS path (gfx1250): guard on builtin existence so we fall back
// to the load+ds_store path if this toolchain doesn't declare them.
// Probe result (round 2): builtin exists; param0 is addrspace(1) int* (printed
// as "__device__ int *"), so cast through non-const int pointers.
#if defined(__AMDGCN__) && \
    __has_builtin(__builtin_amdgcn_global_load_async_to_lds_b32) && \
    __has_builtin(__builtin_amdgcn_s_wait_asynccnt)
#define USE_ASYNC_LDS 1
typedef __attribute__((address_space(1))) int* gptr_t;
typedef __attribute__((address_space(3))) int* lptr_t;
#endif

__global__ __launch_bounds__(512)
void qac_step(const float* __restrict__ src, float* __restrict__ dst,
              const float* __restrict__ attr, const float* __restrict__ lapk,
              int firstStep)
{
    __shared__ float tile[CCH * PLANE];   // halo tile: 16 ch x 18 x 18
    __shared__ float attS[16 * 16];       // attractors zero-padded to 16x16 [k][c]
    __shared__ float anormS[16];          // ||a_k||^2

    const int tid = threadIdx.x;
    const int b   = blockIdx.z;
    const int x0  = blockIdx.x * TW;
    const int y0  = blockIdx.y * TH;

#ifdef USE_ASYNC_LDS
    // Boundary tiles need zero padding: pre-zero LDS only there (uniform branch).
    const bool boundary = (x0 == 0) || (y0 == 0) || (x0 + TW == W_) || (y0 + TH == H_);
    if (boundary) {
        for (int i = tid; i < CCH * PLANE; i += 512) tile[i] = 0.0f;
        __syncthreads();
    }
#endif

    // Stage zero-padded attractor matrix (rows 4..15 = 0).
    if (tid < 256) {
        int k = tid >> 4, c = tid & 15;
        attS[tid] = (k < 4) ? attr[k * 16 + c] : 0.0f;
    }
    if (tid < 16) {
        float s = 0.0f;
        if (tid < 4) {
            for (int c = 0; c < 16; ++c) { float a = attr[tid * 16 + c]; s += a * a; }
        }
        anormS[tid] = s;
    }

    // Cooperative halo fill (16 ch x 18 x 18), zero outside the image.
    for (int i = tid; i < CCH * PLANE; i += 512) {
        int c   = i / PLANE;
        int rem = i - c * PLANE;
        int hy  = rem / HT;
        int hx  = rem - hy * HT;
        int gy  = y0 - 1 + hy;
        int gx  = x0 - 1 + hx;
        bool inb = ((unsigned)gy < H_) && ((unsigned)gx < W_);
#ifdef USE_ASYNC_LDS
        if (inb) {
            const float* g = &src[(((size_t)b * CCH + c) * H_ + gy) * W_ + gx];
            __builtin_amdgcn_global_load_async_to_lds_b32((gptr_t)g, (lptr_t)&tile[i], 0, 0);
        }
#else
        float v = 0.0f;
        if (inb) {
            v = src[(((size_t)b * CCH + c) * H_ + gy) * W_ + gx];
            if (firstStep) v = fminf(fmaxf(v, -2.0f), 2.0f);
        }
        tile[i] = v;
#endif
    }

    // Depthwise 3x3 weights (identical across channels in the reference kernel).
    float w9[9];
#pragma unroll
    for (int t = 0; t < 9; ++t) w9[t] = lapk[t];

#ifdef USE_ASYNC_LDS
    __builtin_amdgcn_s_wait_asynccnt(0);   // my async writes to LDS have landed
    __syncthreads();                        // everyone's have
    if (firstStep) {                        // apply input clip in-place (launch 0 only)
        for (int i = tid; i < CCH * PLANE; i += 512)
            tile[i] = fminf(fmaxf(tile[i], -2.0f), 2.0f);
        __syncthreads();
    }
#else
    __syncthreads();
#endif

    const int lane = tid & 31;
    const int wv   = tid >> 5;      // wave id == tile row (0..15)
    const int px   = lane & 15;     // pixel x within tile / N index
    const int half = lane >> 4;     // fragment half select

    const int hy = wv + 1;
    const int hx = px + 1;

    // z in WMMA C/D layout (VGPR j: lane<16 -> ch j, lane>=16 -> ch j+8) + Laplacian.
    float zc[8], lp[8];
    const int baseZ = (half * 8) * PLANE + (hy - 1) * HT + (hx - 1);
#pragma unroll
    for (int j = 0; j < 8; ++j) {
        const float* p = &tile[baseZ + j * PLANE];
        float t00 = p[0],        t01 = p[1],          t02 = p[2];
        float t10 = p[HT],       t11 = p[HT + 1],     t12 = p[HT + 2];
        float t20 = p[2 * HT],   t21 = p[2 * HT + 1], t22 = p[2 * HT + 2];
        zc[j] = t11;
        lp[j] = w9[0] * t00 + w9[1] * t01 + w9[2] * t02
              + w9[3] * t10 + w9[4] * t11 + w9[5] * t12
              + w9[6] * t20 + w9[7] * t21 + w9[8] * t22;
    }

    // ||z||^2 per pixel: per-lane partial over 8 channels + cross-half add.
    float sp = 0.0f;
#pragma unroll
    for (int j = 0; j < 8; ++j) sp += zc[j] * zc[j];
    float s2 = sp + __shfl_xor(sp, 16);

    // WMMA #1: dots[k][p] = sum_c att[k][c] * z[c][p]  (4 chained K=4 chunks).
    v8f dacc = {0.f, 0.f, 0.f, 0.f, 0.f, 0.f, 0.f, 0.f};
#pragma unroll
    for (int q = 0; q < 4; ++q) {
        v2f af, bf;
        // A (16x4): M=attractor row, K=channel chunk; lanes>=16 hold K+2..K+3.
        af.x = attS[px * 16 + 4 * q + half * 2 + 0];
        af.y = attS[px * 16 + 4 * q + half * 2 + 1];
        // B (4x16): K=channel, N=pixel; loaded straight from the LDS tile.
        int cB = 4 * q + half * 2;
        bf.x = tile[(cB + 0) * PLANE + hy * HT + hx];
        bf.y = tile[(cB + 1) * PLANE + hy * HT + hx];
        dacc = __builtin_amdgcn_wmma_f32_16x16x4_f32(false, af, false, bf,
                                                     (short)0, dacc, false, false);
    }

    // Distances + softmax over 4 attractors (valid in lanes 0..15; upper half benign).
    float l0, l1, l2, l3;
    {
        float d2;
        d2 = s2 + anormS[0] - 2.f * dacc[0]; l0 = -BETA_ * sqrtf(fmaxf(d2, 0.f));
        d2 = s2 + anormS[1] - 2.f * dacc[1]; l1 = -BETA_ * sqrtf(fmaxf(d2, 0.f));
        d2 = s2 + anormS[2] - 2.f * dacc[2]; l2 = -BETA_ * sqrtf(fmaxf(d2, 0.f));
        d2 = s2 + anormS[3] - 2.f * dacc[3]; l3 = -BETA_ * sqrtf(fmaxf(d2, 0.f));
    }
    float m  = fmaxf(fmaxf(l0, l1), fmaxf(l2, l3));
    float e0 = __expf(l0 - m), e1 = __expf(l1 - m);
    float e2 = __expf(l2 - m), e3 = __expf(l3 - m);
    float rs = 1.0f / (e0 + e1 + e2 + e3);
    float w0 = e0 * rs, w1 = e1 * rs, w2 = e2 * rs, w3 = e3 * rs;

    // Broadcast weights into B-fragment layout (K=4 x N=16).
    float g0a = __shfl(w0, px), g0b = __shfl(w2, px);
    float g1a = __shfl(w1, px), g1b = __shfl(w3, px);
    v2f bw; bw.x = half ? g0b : g0a; bw.y = half ? g1b : g1a;

    // WMMA #2: target[c][p] = sum_k attT[c][k] * w[k][p]; D layout matches zc[].
    v2f aT;
    aT.x = attS[(half * 2 + 0) * 16 + px];
    aT.y = attS[(half * 2 + 1) * 16 + px];
    v8f tgt = {0.f, 0.f, 0.f, 0.f, 0.f, 0.f, 0.f, 0.f};
    tgt = __builtin_amdgcn_wmma_f32_16x16x4_f32(false, aT, false, bw,
                                                (short)0, tgt, false, false);

    // Euler update + clamp + store.
    const int gy = y0 + wv;
    const int gx = x0 + px;
#pragma unroll
    for (int j = 0; j < 8; ++j) {
        int   c     = j + half * 8;
        float z     = zc[j];
        float diff  = tgt[j] - z;
        float react = diff * (1.0f - diff * diff);
        float v     = z + DT_ * (DIFF_ * lp[j] + react);
        v = fminf(fmaxf(v, -3.0f), 3.0f);
        dst[(((size_t)b * CCH + c) * H_ + gy) * W_ + gx] = v;
    }
}

extern "C" void kernel_launch(void* const* d_in, const int* in_sizes, int n_in,
                              void* d_out, int out_size, void* d_ws, size_t ws_size,
                              hipStream_t stream)
{
    (void)in_sizes; (void)n_in; (void)out_size; (void)ws_size;
    const float* z    = (const float*)d_in[0];
    const float* attr = (const float*)d_in[1];
    const float* lapk = (const float*)d_in[2];
    float* out = (float*)d_out;
    float* ws  = (float*)d_ws;   // requires >= 32MB scratch (2*16*512*512 floats)

    dim3 grid(W_ / TW, H_ / TH, 2);
    dim3 block(512);

    const float* srcp = z;       // step 0 reads input (with clip to [-2,2])
    for (int s = 0; s < 50; ++s) {
        float* dstp = (s & 1) ? out : ws;   // odd steps -> out; step 49 ends in out
        qac_step<<<grid, block, 0, stream>>>(srcp, dstp, attr, lapk, (s == 0) ? 1 : 0);
        srcp = dstp;
    }
}